// RTMDet_86053964742951
// MI455X (gfx1250) — compile-verified
//
#include <hip/hip_runtime.h>
#include <stdint.h>

#define B_IMG     32
#define NCLS      80
#define NANCH     8400
#define PRE_NMS   1000
#define PRE_PAD   1024
#define MAX_DET   100
#define IOU_TH    0.5f
#define CLS_OFF   4096.0f
#define SORT_N    16384

typedef __attribute__((ext_vector_type(2))) float v2f;
typedef __attribute__((ext_vector_type(8))) float v8f;
typedef int v4i_vs __attribute__((vector_size(16)));   // matches builtin param type

#define AS1 __attribute__((address_space(1)))
#define AS3 __attribute__((address_space(3)))

#if __has_builtin(__builtin_amdgcn_global_load_async_to_lds_b128)
#define HAVE_ASYNC_COPY 1
#endif

// ---------------------------------------------------------------------------
// Kernel A: decode. One thread per (image, anchor). Memory-bound streaming
// (~90 MB total -> ~4us floor at 23.3 TB/s). Class loads are coalesced
// (threads map to W, class stride = H*W). sigmoid is monotone, so
// max/argmax over logits == max/argmax over sigmoid.
// ---------------------------------------------------------------------------
__global__ void decode_kernel(const float* __restrict__ cls0, const float* __restrict__ box0,
                              const float* __restrict__ cls1, const float* __restrict__ box1,
                              const float* __restrict__ cls2, const float* __restrict__ box2,
                              float* __restrict__ boxes, float* __restrict__ scores,
                              int* __restrict__ classes) {
  int t = blockIdx.x * blockDim.x + threadIdx.x;
  if (t >= B_IMG * NANCH) return;
  int b = t / NANCH, a = t - b * NANCH;

  const float* cls; const float* box; int S, step, local;
  if (a < 6400)      { cls = cls0; box = box0; S = 80; step = 8;  local = a; }
  else if (a < 8000) { cls = cls1; box = box1; S = 40; step = 16; local = a - 6400; }
  else               { cls = cls2; box = box2; S = 20; step = 32; local = a - 8000; }

  int h = local / S, w = local - h * S;
  size_t HW = (size_t)S * S;
  const float* cp = cls + (size_t)b * NCLS * HW + (size_t)h * S + w;
  __builtin_prefetch(cp + 16 * HW, 0, 1);   // global_prefetch_b8
  float best = cp[0]; int bi = 0;
  #pragma unroll 4
  for (int c = 1; c < NCLS; ++c) {
    float v = cp[(size_t)c * HW];
    if (v > best) { best = v; bi = c; }
  }
  float conf = 1.0f / (1.0f + __expf(-best));

  const float* bp = box + (size_t)b * 4 * HW + (size_t)h * S + w;
  float dl = bp[0], dt = bp[HW], dr = bp[2 * HW], db = bp[3 * HW];
  float gx = (float)(w * step), gy = (float)(h * step);

  size_t o = (size_t)b * NANCH + a;
  boxes[o * 4 + 0] = gx - dl;
  boxes[o * 4 + 1] = gy - dt;
  boxes[o * 4 + 2] = gx + dr;
  boxes[o * 4 + 3] = gy + db;
  scores[o]  = conf;
  classes[o] = bi;
}

// ---------------------------------------------------------------------------
// Kernel B: per-image full bitonic sort (descending) of 16384 packed keys in
// 128 KB LDS (CDNA5 WGP has 320 KB). key = score_bits<<32 | ~index so equal
// scores break ties toward the smaller index, matching jax top_k.
// ---------------------------------------------------------------------------
__global__ __launch_bounds__(1024) void topk_sort_kernel(const float* __restrict__ scores,
                                                         int* __restrict__ topIdx,
                                                         float* __restrict__ topScr) {
  __shared__ unsigned long long keys[SORT_N];
  int b = blockIdx.x, tid = threadIdx.x;
  const float* sp = scores + (size_t)b * NANCH;

  for (int i = tid; i < SORT_N; i += 1024) {
    unsigned long long k = 0ull;
    if (i < NANCH) {
      unsigned sb = __float_as_uint(sp[i]);            // sigmoid > 0: bits monotone
      k = ((unsigned long long)sb << 32) | (unsigned)(0xFFFFFFFFu - (unsigned)i);
    }
    keys[i] = k;
  }
  __syncthreads();

  for (int k = 2; k <= SORT_N; k <<= 1) {
    for (int j = k >> 1; j > 0; j >>= 1) {
      for (int i = tid; i < SORT_N; i += 1024) {
        int l = i ^ j;
        if (l > i) {
          unsigned long long a = keys[i], c = keys[l];
          if ((a < c) == ((i & k) == 0)) { keys[i] = c; keys[l] = a; }  // descending
        }
      }
      __syncthreads();
    }
  }

  for (int p = tid; p < PRE_NMS; p += 1024) {
    unsigned long long kk = keys[p];
    topIdx[(size_t)b * PRE_NMS + p] = (int)(0xFFFFFFFFu - (unsigned)(kk & 0xFFFFFFFFull));
    topScr[(size_t)b * PRE_NMS + p] = __uint_as_float((unsigned)(kk >> 32));
  }
}

// ---------------------------------------------------------------------------
// Kernel C: NMS. One 1024-thread workgroup (32 wave32) per image.
//  - async-gather the top-1000 boxes straight into LDS (ASYNCcnt path),
//    then add per-class offsets in place
//  - build the IoU>0.5 bit-mask matrix in LDS tile-by-tile; the denominator
//    term S per 16x16 tile is area_M only (area_N sits in B's K=2 row, A's is 0), done
//    with V_WMMA_F32_16X16X4_F32; the threshold test is division-free:
//    inter/(S-inter+eps) > TH  <=>  (1+TH)*inter > TH*(S+eps)   (S+eps > 0)
//  - single wave32 does the serial greedy suppression over 1000 rows
//    (lane = one 32-bit suppression word, shfl-broadcast of the current bit)
//  - thread 0 emits the first 100 survivors (already score-sorted)
// Static LDS: 16K (boxes) + 4K (areas) + 125K (masks) + 128B ~= 148.5 KB,
// so two workgroups still fit per 320 KB WGP.
// ---------------------------------------------------------------------------
__global__ __launch_bounds__(1024) void nms_kernel(const float* __restrict__ boxes,
                                                   const float* __restrict__ scores,
                                                   const int* __restrict__ classes,
                                                   const int* __restrict__ topIdx,
                                                   const float* __restrict__ topScr,
                                                   float* __restrict__ out) {
  __shared__ float4   bx[PRE_PAD];
  __shared__ float    area[PRE_PAD];
  __shared__ unsigned maskLds[PRE_NMS * 32];
  __shared__ unsigned suppLds[32];

  int b = blockIdx.x, tid = threadIdx.x;
  const int* ti = topIdx + (size_t)b * PRE_NMS;

  for (int i = tid; i < PRE_NMS * 32; i += 1024) maskLds[i] = 0u;

  // gather raw boxes for the top-1000 into LDS (async DMA path when available)
  int p = tid;           // PRE_PAD == blockDim.x: one element per thread
  int g = 0;
  if (p < PRE_NMS) {
    g = ti[p];
    const float* src = boxes + ((size_t)b * NANCH + g) * 4;
#ifdef HAVE_ASYNC_COPY
    __builtin_amdgcn_global_load_async_to_lds_b128(
        (AS1 v4i_vs*)(uintptr_t)src,
        (AS3 v4i_vs*)(uintptr_t)&bx[p], 0, 0);
#else
    bx[p] = *(const float4*)src;
#endif
  }
#ifdef HAVE_ASYNC_COPY
#if __has_builtin(__builtin_amdgcn_s_wait_asynccnt)
  __builtin_amdgcn_s_wait_asynccnt(0);
#else
  asm volatile("s_wait_asynccnt 0x0" ::: "memory");
#endif
#endif
  __syncthreads();

  // apply class offsets in place, compute areas, fill padding
  {
    float4 v; float ar;
    if (p < PRE_NMS) {
      float off = (float)classes[(size_t)b * NANCH + g] * CLS_OFF;
      v = bx[p];
      v.x += off; v.y += off; v.z += off; v.w += off;
      ar = fmaxf(v.z - v.x, 0.f) * fmaxf(v.w - v.y, 0.f);
    } else {
      float f = -1.0e9f - (float)p * 1.0e4f;   // degenerate far box: IoU == 0
      v = make_float4(f, f, f, f);
      ar = 0.f;
    }
    bx[p] = v; area[p] = ar;
  }
  __syncthreads();

  // 64x64 tiles of 16x16; uniform per-wave loop -> EXEC all-1s at the WMMA
  int wave = tid >> 5, lane = tid & 31;
  for (int tile = wave; tile < 64 * 64; tile += 32) {
    int m0 = (tile >> 6) * 16, n0 = (tile & 63) * 16;

    v2f a, bb; v8f acc = {};
    // A (16x4): lanes 0-15 hold K=0,1 -> [area_M, 1]; lanes 16-31 hold K=2,3 -> 0
    a.x  = (lane < 16) ? area[m0 + (lane & 15)] : 0.f;
    a.y  = (lane < 16) ? 1.0f : 0.f;
    // B (4x16): K=0 row = ones, K=2 row = area_N, K=1,3 rows = 0
    bb.x = (lane < 16) ? 1.0f : area[n0 + (lane & 15)];
    bb.y = 0.f;
    acc = __builtin_amdgcn_wmma_f32_16x16x4_f32(false, a, false, bb,
                                                (short)0, acc, false, false);
    // acc[r] = area[M] only (area[N] sits in K=2, where A is 0) per the 16x16 f32 C/D layout
    int N = n0 + (lane & 15);
    float4 bn = bx[N];
    #pragma unroll
    for (int r = 0; r < 8; ++r) {
      int M = m0 + r + ((lane >= 16) ? 8 : 0);
      float4 bm = bx[M];
      float iw = fmaxf(fminf(bm.z, bn.z) - fmaxf(bm.x, bn.x), 0.f);
      float ih = fmaxf(fminf(bm.w, bn.w) - fmaxf(bm.y, bn.y), 0.f);
      float inter = iw * ih;
      // division-free IoU threshold (denominator strictly positive)
      bool hit = (1.0f + IOU_TH) * inter > IOU_TH * (acc[r] + 1e-7f);
      if (hit && N > M && M < PRE_NMS && N < PRE_NMS)
        atomicOr(&maskLds[M * 32 + (N >> 5)], 1u << (N & 31));   // ds_or_b32
    }
  }
  __syncthreads();

  // serial greedy pass: one wave32, lane owns suppression word `lane`
  if (wave == 0) {
    unsigned supp = 0u;
    for (int i = 0; i < PRE_NMS; ++i) {
      unsigned cw = __shfl(supp, i >> 5, 32);
      if (!((cw >> (i & 31)) & 1u))
        supp |= maskLds[i * 32 + lane];
    }
    suppLds[lane] = supp;
  }
  __syncthreads();

  // emit first 100 survivors (already score-sorted); pad with zeros / class -1
  if (tid == 0) {
    float* outB = out;                                  // [B,100,5] f32
    int*   outC = (int*)(out + B_IMG * MAX_DET * 5);    // [B,100]   i32 bits
    int kept = 0;
    for (int i = 0; i < PRE_NMS && kept < MAX_DET; ++i) {
      if ((suppLds[i >> 5] >> (i & 31)) & 1u) continue;
      int gi = ti[i];
      size_t o  = ((size_t)b * NANCH + gi) * 4;
      size_t ob = (size_t)b * MAX_DET + kept;
      outB[ob * 5 + 0] = boxes[o + 0];
      outB[ob * 5 + 1] = boxes[o + 1];
      outB[ob * 5 + 2] = boxes[o + 2];
      outB[ob * 5 + 3] = boxes[o + 3];
      outB[ob * 5 + 4] = topScr[(size_t)b * PRE_NMS + i];
      outC[ob] = classes[(size_t)b * NANCH + gi];
      ++kept;
    }
    for (; kept < MAX_DET; ++kept) {
      size_t ob = (size_t)b * MAX_DET + kept;
      outB[ob * 5 + 0] = 0.f; outB[ob * 5 + 1] = 0.f; outB[ob * 5 + 2] = 0.f;
      outB[ob * 5 + 3] = 0.f; outB[ob * 5 + 4] = 0.f;
      outC[ob] = -1;
    }
  }
}

// ---------------------------------------------------------------------------
extern "C" void kernel_launch(void* const* d_in, const int* in_sizes, int n_in,
                              void* d_out, int out_size, void* d_ws, size_t ws_size,
                              hipStream_t stream) {
  const float* cls0 = (const float*)d_in[0];
  const float* box0 = (const float*)d_in[1];
  const float* cls1 = (const float*)d_in[2];
  const float* box1 = (const float*)d_in[3];
  const float* cls2 = (const float*)d_in[4];
  const float* box2 = (const float*)d_in[5];

  char* ws = (char*)d_ws;
  float* boxes   = (float*)ws;  ws += (size_t)B_IMG * NANCH * 4 * sizeof(float);
  float* scores  = (float*)ws;  ws += (size_t)B_IMG * NANCH * sizeof(float);
  int*   classes = (int*)ws;    ws += (size_t)B_IMG * NANCH * sizeof(int);
  int*   topIdx  = (int*)ws;    ws += (size_t)B_IMG * PRE_NMS * sizeof(int);
  float* topScr  = (float*)ws;

  int total = B_IMG * NANCH;
  decode_kernel<<<(total + 255) / 256, 256, 0, stream>>>(cls0, box0, cls1, box1,
                                                         cls2, box2, boxes, scores, classes);
  topk_sort_kernel<<<B_IMG, 1024, 0, stream>>>(scores, topIdx, topScr);
  nms_kernel<<<B_IMG, 1024, 0, stream>>>(boxes, scores, classes, topIdx, topScr,
                                         (float*)d_out);
}